// Distribute_43284680409183
// MI455X (gfx1250) — compile-verified
//
#include <hip/hip_runtime.h>

// Per-pixel locally-connected MLP, N = [16,16,3], weight [B=4, K=337, H=256, W=256].
// Memory-bound (0.47 FLOP/byte): streaming kernel on the CDNA5 async global->LDS
// pipeline (ASYNCcnt double buffering), per-lane FMA compute.
//
// All accumulator indices are template parameters so they are IR constants before
// SROA runs -> accumulators stay in VGPRs (no scratch spills).

#define GLOBAL_AS __attribute__((address_space(1)))
#define LDS_AS    __attribute__((address_space(3)))

typedef int v2i __attribute__((ext_vector_type(2)));

constexpr int KTOT   = 337;        // 16*2 + 16*16 + 3*16 + 1 bias channel
constexpr int HW     = 256 * 256;  // pixels per image
constexpr int CH     = 8;          // channels per async chunk
constexpr int NCHUNK = 43;         // 42*8 + 1 = 337
constexpr int LANES  = 32;         // wave32
constexpr int WAVES  = 8;          // 256-thread block

__device__ __forceinline__ void async_copy_b64(const float* gsrc, float2* ldst) {
#if __has_builtin(__builtin_amdgcn_global_load_async_to_lds_b64)
  __builtin_amdgcn_global_load_async_to_lds_b64(
      (GLOBAL_AS v2i*)gsrc, (LDS_AS v2i*)ldst, /*offset=*/0, /*cpol=*/0);
#else
  unsigned lds_off = (unsigned)(unsigned long long)ldst;
  asm volatile("global_load_async_to_lds_b64 %0, %1, off"
               :: "v"(lds_off), "v"((unsigned long long)gsrc)
               : "memory");
#endif
}

template <int N>
__device__ __forceinline__ void wait_async() {
#if __has_builtin(__builtin_amdgcn_s_wait_asynccnt)
  __builtin_amdgcn_s_wait_asynccnt(N);
#else
  asm volatile("s_wait_asynccnt %0" :: "n"(N) : "memory");
#endif
  asm volatile("" ::: "memory");  // keep LDS reads below the wait
}

struct State {
  float2 h1[16];
  float2 h2[16];
  float2 h3[3];
  float2 biasv;
};

// ---- template helpers: every index is an IR constant -----------------------

template <int I>
__device__ __forceinline__ void zero16(float2 (&a)[16]) {
  if constexpr (I < 16) { a[I] = make_float2(0.f, 0.f); zero16<I + 1>(a); }
}

template <int I>
__device__ __forceinline__ void relu16(float2 (&a)[16]) {
  if constexpr (I < 16) {
    a[I].x = fmaxf(a[I].x, 0.f);
    a[I].y = fmaxf(a[I].y, 0.f);
    relu16<I + 1>(a);
  }
}

template <int K>
__device__ __forceinline__ void apply_channel(State& s, float2 w, float c0, float c1) {
  if constexpr (K < 32) {                    // layer 0: k = n*2 + d
    constexpr int n = K >> 1;
    const float cc = (K & 1) ? c1 : c0;
    s.h1[n].x = fmaf(w.x, cc, s.h1[n].x);
    s.h1[n].y = fmaf(w.y, cc, s.h1[n].y);
  } else if constexpr (K < 288) {            // layer 1: k = 32 + m*16 + n
    constexpr int m = (K - 32) >> 4, n = (K - 32) & 15;
    s.h2[m].x = fmaf(w.x, s.h1[n].x, s.h2[m].x);
    s.h2[m].y = fmaf(w.y, s.h1[n].y, s.h2[m].y);
  } else if constexpr (K < 336) {            // layer 2: k = 288 + j*16 + n
    constexpr int j = (K - 288) >> 4, n = (K - 288) & 15;
    s.h3[j].x = fmaf(w.x, s.h2[n].x, s.h3[j].x);
    s.h3[j].y = fmaf(w.y, s.h2[n].y, s.h3[j].y);
  } else {                                   // k == 336: bias channel
    s.biasv = w;
  }
  if constexpr (K == 31)  relu16<0>(s.h1);   // ReLU after layer 0
  if constexpr (K == 287) relu16<0>(s.h2);   // ReLU after layer 1
}

template <int C, int CHI>
__device__ __forceinline__ void consume_chunk(State& s, const float2 (&buf)[CH][LANES],
                                              int lane, float c0, float c1) {
  if constexpr (CHI < CH && (C * CH + CHI) < KTOT) {
    float2 w = buf[CHI][lane];               // LDS read (real memory; dynamic lane OK)
    apply_channel<C * CH + CHI>(s, w, c0, c1);
    consume_chunk<C, CHI + 1>(s, buf, lane, c0, c1);
  }
}

template <int C, int CHI>
__device__ __forceinline__ void issue_chunk(const float* wbase, float2 (&buf)[CH][LANES],
                                            int lane) {
  if constexpr (CHI < CH && (C * CH + CHI) < KTOT) {
    async_copy_b64(wbase + (size_t)(C * CH + CHI) * HW, &buf[CHI][lane]);
    issue_chunk<C, CHI + 1>(wbase, buf, lane);
  }
}

template <int C>
__device__ __forceinline__ void pipeline(State& s, const float* wbase,
                                         float2 (&stg)[2][CH][LANES],
                                         int lane, float c0, float c1) {
  if constexpr (C < NCHUNK) {
    if constexpr (C + 1 < NCHUNK) {
      issue_chunk<C + 1, 0>(wbase, stg[(C + 1) & 1], lane);
      // async loads complete in order: ASYNCcnt <= pending(next chunk)  ==>  chunk C landed
      constexpr int pending = (C + 1 == NCHUNK - 1) ? (KTOT - (NCHUNK - 1) * CH) : CH;
      wait_async<pending>();
    } else {
      wait_async<0>();
    }
    consume_chunk<C, 0>(s, stg[C & 1], lane, c0, c1);
    pipeline<C + 1>(s, wbase, stg, lane, c0, c1);
  }
}

// ---- kernel ----------------------------------------------------------------

__global__ void __launch_bounds__(256)
pixel_mlp_kernel(const float* __restrict__ weight,
                 const float* __restrict__ coor,
                 float* __restrict__ out) {
  // Per-wave double-buffered staging: 8 waves * 2 bufs * 8 ch * 32 lanes * 8B = 32 KB
  __shared__ float2 stage[WAVES][2][CH][LANES];

  const int tid  = (int)threadIdx.x;
  const int wv   = tid >> 5;
  const int lane = tid & 31;
  const int blk  = (int)blockIdx.x;                  // 512 blocks total
  const int b    = blk >> 7;                         // image index 0..3
  const int pp   = ((blk & 127) << 9) + (tid << 1);  // pixel offset in image (2 px/thread)

  const float c0 = coor[0];
  const float c1 = coor[1];

  const float* wbase = weight + (size_t)b * KTOT * HW + pp;

  State s;
  zero16<0>(s.h1);
  zero16<0>(s.h2);
  s.h3[0] = make_float2(0.f, 0.f);
  s.h3[1] = make_float2(0.f, 0.f);
  s.h3[2] = make_float2(0.f, 0.f);
  s.biasv = make_float2(0.f, 0.f);

  issue_chunk<0, 0>(wbase, stage[wv][0], lane);      // prologue
  pipeline<0>(s, wbase, stage[wv], lane, c0, c1);    // double-buffered steady state

  // out[b, j, h, w] = h3[j] + bias   (constant j indices)
  float* obase = out + (size_t)(b * 3) * HW + pp;
  *reinterpret_cast<float2*>(obase + 0 * (size_t)HW) =
      make_float2(s.h3[0].x + s.biasv.x, s.h3[0].y + s.biasv.y);
  *reinterpret_cast<float2*>(obase + 1 * (size_t)HW) =
      make_float2(s.h3[1].x + s.biasv.x, s.h3[1].y + s.biasv.y);
  *reinterpret_cast<float2*>(obase + 2 * (size_t)HW) =
      make_float2(s.h3[2].x + s.biasv.x, s.h3[2].y + s.biasv.y);
}

extern "C" void kernel_launch(void* const* d_in, const int* in_sizes, int n_in,
                              void* d_out, int out_size, void* d_ws, size_t ws_size,
                              hipStream_t stream) {
  // setup_inputs order: d_in[0] = input (1 elem, unused by forward),
  //                     d_in[1] = weight [4, 337, 256, 256] fp32,
  //                     d_in[2] = coor [2] fp32
  const float* weight = (const float*)d_in[1];
  const float* coor   = (const float*)d_in[2];
  float*       out    = (float*)d_out;  // [4, 3, 256, 256] fp32

  // 262144 pixels, 2 px/thread -> 131072 threads = 512 blocks x 256 threads
  hipLaunchKernelGGL(pixel_mlp_kernel, dim3(512), dim3(256), 0, stream,
                     weight, coor, out);

  (void)d_in; (void)in_sizes; (void)n_in; (void)out_size; (void)d_ws; (void)ws_size;
}